// EEGGAT_62663572849445
// MI455X (gfx1250) — compile-verified
//
#include <hip/hip_runtime.h>
#include <math.h>

#define FDIM 128
#define HID 32
#define HEADS 4

typedef __attribute__((ext_vector_type(2))) float v2f;
typedef __attribute__((ext_vector_type(8))) float v8f;

__device__ __forceinline__ void atomicMaxF(float* addr, float val) {
  // works because floats compare like ints when >=0 (signed max) and like
  // reversed uints when <0 (unsigned min); init value -3.4e38 is valid in both.
  if (val >= 0.0f) atomicMax((int*)addr, __float_as_int(val));
  else             atomicMin((unsigned int*)addr, __float_as_uint(val));
}

// -------------------- WMMA GEMM: Y[n x 128] = X[n x 128] @ W[128 x 128] ----
__global__ void gat_gemm128(const float* __restrict__ X,
                            const float* __restrict__ W,
                            float* __restrict__ Y, int n) {
  const int wave = threadIdx.x >> 5;
  const int lane = threadIdx.x & 31;
  const int tile = blockIdx.x * (blockDim.x >> 5) + wave;
  const int row_tiles = (n + 15) >> 4;
  if (tile >= row_tiles * (FDIM / 16)) return;
  const int tm = tile >> 3;   // row tile (FDIM/16 == 8 col tiles)
  const int tn = tile & 7;    // col tile
  const int l16 = lane & 15;
  const int hi  = lane >> 4;  // 0: K+{0,1}, 1: K+{2,3}

  int arow = tm * 16 + l16;
  if (arow >= n) arow = n - 1;            // clamp (stores are guarded)
  const float* xp = X + (long)arow * FDIM;
  const int bcol = tn * 16 + l16;

  v8f c = {};
  for (int k = 0; k < FDIM; k += 4) {
    v2f a, b;
    a.x = xp[k + 2 * hi];
    a.y = xp[k + 2 * hi + 1];
    b.x = W[(k + 2 * hi) * FDIM + bcol];
    b.y = W[(k + 2 * hi + 1) * FDIM + bcol];
    c = __builtin_amdgcn_wmma_f32_16x16x4_f32(false, a, false, b,
                                              (short)0, c, false, false);
  }
#pragma unroll
  for (int r = 0; r < 8; ++r) {
    const int row = tm * 16 + r + 8 * hi;
    if (row < n) Y[(long)row * FDIM + tn * 16 + l16] = c[r];
  }
}

// -------------------- fill -------------------------------------------------
__global__ void gat_fill(float* __restrict__ p, float v, long long n) {
  long long i = (long long)blockIdx.x * blockDim.x + threadIdx.x;
  if (i < n) p[i] = v;
}

// -------------------- per-node attention logits ----------------------------
__global__ void gat_node_logits(const float* __restrict__ h,
                                const float* __restrict__ a_src,
                                const float* __restrict__ a_dst,
                                float* __restrict__ als,
                                float* __restrict__ ald, int n) {
  int i = blockIdx.x * blockDim.x + threadIdx.x;   // over n*HEADS
  if (i >= n * HEADS) return;
  const int node = i >> 2, hd = i & 3;
  const float* hp = h + (long)node * FDIM + hd * HID;
  const float* as = a_src + hd * HID;
  const float* ad = a_dst + hd * HID;
  float s = 0.f, d = 0.f;
#pragma unroll
  for (int f = 0; f < HID; ++f) { float v = hp[f]; s += v * as[f]; d += v * ad[f]; }
  als[i] = s; ald[i] = d;
}

// -------------------- per-edge logits + segment max ------------------------
__global__ void gat_edge_logits(const int* __restrict__ src, const int* __restrict__ dst,
                                const float* __restrict__ als, const float* __restrict__ ald,
                                float* __restrict__ eb, float* __restrict__ m,
                                int E, int Etot) {
  int i = blockIdx.x * blockDim.x + threadIdx.x;
  if (i >= Etot) return;
  const int s = (i < E) ? src[i] : (i - E);
  const int d = (i < E) ? dst[i] : (i - E);
#pragma unroll
  for (int h = 0; h < HEADS; ++h) {
    float e = als[s * HEADS + h] + ald[d * HEADS + h];
    e = (e > 0.f) ? e : 0.2f * e;                  // leaky_relu 0.2
    eb[(long)i * HEADS + h] = e;
    atomicMaxF(&m[d * HEADS + h], e);
  }
}

// -------------------- exp + segment sum ------------------------------------
__global__ void gat_edge_exp(const int* __restrict__ dst,
                             const float* __restrict__ m,
                             float* __restrict__ eb, float* __restrict__ den,
                             int E, int Etot) {
  int i = blockIdx.x * blockDim.x + threadIdx.x;
  if (i >= Etot) return;
  const int d = (i < E) ? dst[i] : (i - E);
#pragma unroll
  for (int h = 0; h < HEADS; ++h) {
    float ex = expf(eb[(long)i * HEADS + h] - m[d * HEADS + h]);
    eb[(long)i * HEADS + h] = ex;
    atomicAdd(&den[d * HEADS + h], ex);
  }
}

// -------------------- weighted scatter-add ---------------------------------
__global__ void gat_edge_scatter(const int* __restrict__ src, const int* __restrict__ dst,
                                 const float* __restrict__ eb, const float* __restrict__ den,
                                 const float* __restrict__ hlin, float* __restrict__ hout,
                                 int E, int Etot) {
  long long i = (long long)blockIdx.x * blockDim.x + threadIdx.x;  // over Etot*FDIM
  if (i >= (long long)Etot * FDIM) return;
  const int eidx = (int)(i >> 7);
  const int f = (int)(i & (FDIM - 1));
  const int h = f >> 5;                           // head = f / HID
  const int s = (eidx < E) ? src[eidx] : (eidx - E);
  const int d = (eidx < E) ? dst[eidx] : (eidx - E);
  const float alpha = eb[(long)eidx * HEADS + h] / (den[d * HEADS + h] + 1e-16f);
  atomicAdd(&hout[(long)d * FDIM + f], hlin[(long)s * FDIM + f] * alpha);
}

// -------------------- bias + ELU -------------------------------------------
__global__ void gat_bias_elu(float* __restrict__ h, const float* __restrict__ b, int n) {
  long long i = (long long)blockIdx.x * blockDim.x + threadIdx.x;
  if (i >= (long long)n * FDIM) return;
  float v = h[i] + b[i & (FDIM - 1)];
  h[i] = (v > 0.f) ? v : expm1f(v);               // ELU alpha=1
}

// -------------------- mean pool accumulation -------------------------------
__global__ void gat_pool(const float* __restrict__ h, const int* __restrict__ batch,
                         float* __restrict__ sums, float* __restrict__ cnts, int n) {
  long long i = (long long)blockIdx.x * blockDim.x + threadIdx.x;
  if (i >= (long long)n * FDIM) return;
  const int node = (int)(i >> 7), c = (int)(i & (FDIM - 1));
  const int g = batch[node];
  atomicAdd(&sums[(long)g * FDIM + c], h[i]);
  if (c == 0) atomicAdd(&cnts[g], 1.0f);
}

// -------------------- head MLP (tiny: one block) ---------------------------
__global__ void gat_head(const float* __restrict__ sums, const float* __restrict__ cnts,
                         const float* __restrict__ Wh, const float* __restrict__ bh,
                         const float* __restrict__ Wc, const float* __restrict__ bc,
                         float* __restrict__ out, int Gn) {
  int g = threadIdx.x;
  if (g >= Gn) return;
  const float inv = 1.0f / fmaxf(cnts[g], 1.0f);
  const float* sg = sums + (long)g * FDIM;
  float acc = bc[0];
  for (int j = 0; j < HID; ++j) {
    float z = 0.f;
    for (int c = 0; c < FDIM; ++c) z += sg[c] * Wh[c * HID + j];
    z = z * inv + bh[j];
    z = fmaxf(z, 0.f);                            // ReLU
    acc += z * Wc[j];
  }
  out[g] = acc;
}

// ---------------------------------------------------------------------------
static void run_gat_layer(const float* hin, const float* W, const float* a_s,
                          const float* a_d, const float* bias,
                          float* hlin, float* hout,
                          float* eb, float* als, float* ald, float* m, float* den,
                          const int* srcp, const int* dstp,
                          int N, int E, int Etot, hipStream_t stream) {
  const int row_tiles = (N + 15) >> 4;
  const int tiles = row_tiles * (FDIM / 16);
  gat_gemm128<<<(tiles + 3) / 4, 128, 0, stream>>>(hin, W, hlin, N);

  gat_node_logits<<<(N * HEADS + 255) / 256, 256, 0, stream>>>(hlin, a_s, a_d, als, ald, N);

  gat_fill<<<((long long)N * HEADS + 255) / 256, 256, 0, stream>>>(m, -3.4e38f, (long long)N * HEADS);
  gat_fill<<<((long long)N * HEADS + 255) / 256, 256, 0, stream>>>(den, 0.f, (long long)N * HEADS);
  gat_fill<<<((long long)N * FDIM + 255) / 256, 256, 0, stream>>>(hout, 0.f, (long long)N * FDIM);

  gat_edge_logits<<<(Etot + 255) / 256, 256, 0, stream>>>(srcp, dstp, als, ald, eb, m, E, Etot);
  gat_edge_exp<<<(Etot + 255) / 256, 256, 0, stream>>>(dstp, m, eb, den, E, Etot);

  const long long sc = (long long)Etot * FDIM;
  gat_edge_scatter<<<(unsigned)((sc + 255) / 256), 256, 0, stream>>>(srcp, dstp, eb, den,
                                                                    hlin, hout, E, Etot);
  gat_bias_elu<<<(unsigned)(((long long)N * FDIM + 255) / 256), 256, 0, stream>>>(hout, bias, N);
}

extern "C" void kernel_launch(void* const* d_in, const int* in_sizes, int n_in,
                              void* d_out, int out_size, void* d_ws, size_t ws_size,
                              hipStream_t stream) {
  const float* x    = (const float*)d_in[0];
  const int*   ei   = (const int*)d_in[1];     // [2, E]
  const int*   batch= (const int*)d_in[2];
  const float* W1   = (const float*)d_in[3];
  const float* a1s  = (const float*)d_in[4];
  const float* a1d  = (const float*)d_in[5];
  const float* b1   = (const float*)d_in[6];
  const float* W2   = (const float*)d_in[7];
  const float* a2s  = (const float*)d_in[8];
  const float* a2d  = (const float*)d_in[9];
  const float* b2   = (const float*)d_in[10];
  const float* Wh   = (const float*)d_in[11];
  const float* bh   = (const float*)d_in[12];
  const float* Wc   = (const float*)d_in[13];
  const float* bc   = (const float*)d_in[14];
  float* out = (float*)d_out;

  const int N = in_sizes[0] / FDIM;
  const int E = in_sizes[1] / 2;
  const int Etot = E + N;
  const int Gn = out_size;

  // workspace layout (floats)
  float* ws   = (float*)d_ws;
  float* A    = ws;                               // N*128  (h_lin)
  float* B    = A + (size_t)N * FDIM;             // N*128  (h_out / layer input)
  float* eb   = B + (size_t)N * FDIM;             // Etot*HEADS
  float* als  = eb + (size_t)Etot * HEADS;        // N*HEADS
  float* ald  = als + (size_t)N * HEADS;          // N*HEADS
  float* m    = ald + (size_t)N * HEADS;          // N*HEADS
  float* den  = m + (size_t)N * HEADS;            // N*HEADS
  float* sums = den + (size_t)N * HEADS;          // G*128
  float* cnts = sums + (size_t)Gn * FDIM;         // G

  const int* srcp = ei;
  const int* dstp = ei + E;

  // Layer 1: x -> B
  run_gat_layer(x, W1, a1s, a1d, b1, A, B, eb, als, ald, m, den,
                srcp, dstp, N, E, Etot, stream);
  // Layer 2: B -> B (gemm reads B into A first; B is then zeroed & reused)
  run_gat_layer(B, W2, a2s, a2d, b2, A, B, eb, als, ald, m, den,
                srcp, dstp, N, E, Etot, stream);

  // global mean pool + head
  gat_fill<<<((long long)Gn * FDIM + 255) / 256, 256, 0, stream>>>(sums, 0.f, (long long)Gn * FDIM);
  gat_fill<<<(Gn + 255) / 256, 256, 0, stream>>>(cnts, 0.f, (long long)Gn);
  gat_pool<<<(unsigned)(((long long)N * FDIM + 255) / 256), 256, 0, stream>>>(B, batch, sums, cnts, N);
  gat_head<<<1, 128, 0, stream>>>(sums, cnts, Wh, bh, Wc, bc, out, Gn);
}